// MyGCN_edge_38397007626982
// MI455X (gfx1250) — compile-verified
//
#include <hip/hip_runtime.h>
#include <math.h>

typedef float v2f __attribute__((ext_vector_type(2)));
typedef float v8f __attribute__((ext_vector_type(8)));

#define HD 128

// ---------------- GEMM: Hout[nrows x 128] = X[nrows x 128] @ W[128 x 128] ----
// grid.x = nrows/16, block = 256 (8 waves). Wave w computes the 16x16 output
// tile at (rows blockIdx.x*16, cols w*16) via fp32 WMMA, K marched in steps of 4.
__global__ __launch_bounds__(256)
void gemm_wmma_f32(const float* __restrict__ X, const float* __restrict__ W,
                   float* __restrict__ Hout) {
    const int wave = threadIdx.x >> 5;          // 0..7 -> N tile
    const int lane = threadIdx.x & 31;
    const int half = lane >> 4;                 // 0: K pair {0,1}, 1: K pair {2,3}
    const int lr   = lane & 15;
    const int m0   = blockIdx.x * 16;
    const int n0   = wave * 16;

    v8f c = {};
    const float* xrow = X + (size_t)(m0 + lr) * HD;
    #pragma unroll 4
    for (int k0 = 0; k0 < HD; k0 += 4) {
        const int ka = k0 + 2 * half;
        v2f a, b;
        a.x = xrow[ka];
        a.y = xrow[ka + 1];
        b.x = W[(size_t)ka * HD + n0 + lr];
        b.y = W[(size_t)(ka + 1) * HD + n0 + lr];
        c = __builtin_amdgcn_wmma_f32_16x16x4_f32(
                /*neg_a=*/false, a, /*neg_b=*/false, b,
                /*c_mod=*/(short)0, c, /*reuse_a=*/false, /*reuse_b=*/false);
    }
    // D layout: VGPR r, lanes 0-15 -> M=r, N=lane; lanes 16-31 -> M=8+r, N=lane-16
    const int col  = n0 + lr;
    const int mrow = m0 + half * 8;
    #pragma unroll
    for (int r = 0; r < 8; ++r)
        Hout[(size_t)(mrow + r) * HD + col] = c[r];
}

// ---------------- degree / dinv ---------------------------------------------
__global__ void init_deg(float* __restrict__ deg, int n) {
    int i = blockIdx.x * blockDim.x + threadIdx.x;
    if (i < n) deg[i] = 1.0f;                  // self-loop contributes 1
}
__global__ void count_deg(const long long* __restrict__ dst, float* __restrict__ deg, int e) {
    int i = blockIdx.x * blockDim.x + threadIdx.x;
    if (i < e) atomicAdd(&deg[(int)dst[i]], 1.0f);
}
__global__ void rsqrt_deg(float* __restrict__ deg, int n) {
    int i = blockIdx.x * blockDim.x + threadIdx.x;
    if (i < n) deg[i] = rsqrtf(deg[i]);        // in place: deg -> dinv
}

// ---------------- self-loop init: agg = dinv^2 * h ---------------------------
__global__ void selfloop_init(const float* __restrict__ h, const float* __restrict__ dinv,
                              float* __restrict__ agg, int total) {
    int i = blockIdx.x * blockDim.x + threadIdx.x;
    if (i < total) {
        float dv = dinv[i >> 7];
        agg[i] = dv * dv * h[i];
    }
}

// ---------------- edge scatter: agg[dst] += dinv[s]*dinv[d]*h[src] ----------
// One wave per edge: 32 lanes x 4 floats = one 512B row, per-lane f32 atomics.
__global__ __launch_bounds__(256)
void edge_scatter(const long long* __restrict__ src, const long long* __restrict__ dst,
                  const float* __restrict__ dinv, const float* __restrict__ h,
                  float* __restrict__ agg, int nedges) {
    int e = blockIdx.x * 8 + (threadIdx.x >> 5);
    if (e >= nedges) return;
    const int lane = threadIdx.x & 31;
    const int s = (int)src[e];
    const int d = (int)dst[e];
    const float norm = dinv[s] * dinv[d];
    const float* hs = h + (size_t)s * HD;
    float* ad = agg + (size_t)d * HD;
    #pragma unroll
    for (int j = lane; j < HD; j += 32)
        atomicAdd(&ad[j], norm * hs[j]);
}

// ---------------- bias + relu (in place) ------------------------------------
__global__ void bias_relu(float* __restrict__ agg, const float* __restrict__ b, int total) {
    int i = blockIdx.x * blockDim.x + threadIdx.x;
    if (i < total) agg[i] = fmaxf(agg[i] + b[i & (HD - 1)], 0.0f);
}

// ---------------- gate: sigmoid(x . gw + gb) per node (wave per node) -------
__global__ __launch_bounds__(256)
void gate_kernel(const float* __restrict__ x, const float* __restrict__ gw,
                 const float* __restrict__ gb, float* __restrict__ gate, int n) {
    int i = blockIdx.x * 8 + (threadIdx.x >> 5);
    if (i >= n) return;
    const int lane = threadIdx.x & 31;
    const float* xr = x + (size_t)i * HD;
    float acc = 0.0f;
    #pragma unroll
    for (int j = lane; j < HD; j += 32) acc += xr[j] * gw[j];
    #pragma unroll
    for (int off = 16; off > 0; off >>= 1) acc += __shfl_down(acc, off);
    if (lane == 0) gate[i] = 1.0f / (1.0f + __expf(-(acc + gb[0])));
}

// ---------------- softmax stats: stats[0]=max(gate), stats[1]=sum(exp(g-max))
__global__ __launch_bounds__(1024)
void softmax_stats(const float* __restrict__ gate, int n, float* __restrict__ stats) {
    __shared__ float sred[32];
    const int wave = threadIdx.x >> 5, lane = threadIdx.x & 31;

    float m = -1e30f;
    for (int i = threadIdx.x; i < n; i += 1024) m = fmaxf(m, gate[i]);
    #pragma unroll
    for (int off = 16; off > 0; off >>= 1) m = fmaxf(m, __shfl_down(m, off));
    if (lane == 0) sred[wave] = m;
    __syncthreads();
    if (wave == 0) {
        m = sred[lane];
        #pragma unroll
        for (int off = 16; off > 0; off >>= 1) m = fmaxf(m, __shfl_down(m, off));
        if (lane == 0) sred[0] = m;
    }
    __syncthreads();
    const float gm = sred[0];
    __syncthreads();

    float s = 0.0f;
    for (int i = threadIdx.x; i < n; i += 1024) s += __expf(gate[i] - gm);
    #pragma unroll
    for (int off = 16; off > 0; off >>= 1) s += __shfl_down(s, off);
    if (lane == 0) sred[wave] = s;
    __syncthreads();
    if (wave == 0) {
        s = sred[lane];
        #pragma unroll
        for (int off = 16; off > 0; off >>= 1) s += __shfl_down(s, off);
        if (lane == 0) { stats[0] = gm; stats[1] = s; }
    }
}

// ---------------- pooled sum: out[j] += sum_i softmax(g)_i * x[i][j] --------
// 128 threads = one coalesced row per iteration; partials via atomics.
__global__ __launch_bounds__(128)
void pool_kernel(const float* __restrict__ x, const float* __restrict__ gate,
                 const float* __restrict__ stats, float* __restrict__ out, int n) {
    const int j = threadIdx.x;
    const float gm = stats[0];
    const float inv = 1.0f / stats[1];
    float acc = 0.0f;
    for (int i = blockIdx.x; i < n; i += gridDim.x)
        acc += __expf(gate[i] - gm) * inv * x[(size_t)i * HD + j];
    atomicAdd(&out[j], acc);
}

__global__ void zero_kernel(float* __restrict__ p, int n) {
    int i = blockIdx.x * blockDim.x + threadIdx.x;
    if (i < n) p[i] = 0.0f;
}

// ---------------- host-side layer driver ------------------------------------
static void run_layer(const float* X, const float* W, const float* b,
                      const float* gw, const float* gb,
                      const long long* srcp, const long long* dstp,
                      const float* dinv, float* h, float* xout,
                      float* gatebuf, float* stats, float* outseg,
                      int Nn, int Ee, hipStream_t stream) {
    const int NHtot = Nn * HD;
    gemm_wmma_f32<<<Nn / 16, 256, 0, stream>>>(X, W, h);
    selfloop_init<<<(NHtot + 255) / 256, 256, 0, stream>>>(h, dinv, xout, NHtot);
    edge_scatter<<<(Ee + 7) / 8, 256, 0, stream>>>(srcp, dstp, dinv, h, xout, Ee);
    bias_relu<<<(NHtot + 255) / 256, 256, 0, stream>>>(xout, b, NHtot);
    gate_kernel<<<(Nn + 7) / 8, 256, 0, stream>>>(xout, gw, gb, gatebuf, Nn);
    softmax_stats<<<1, 1024, 0, stream>>>(gatebuf, Nn, stats);
    pool_kernel<<<128, 128, 0, stream>>>(xout, gatebuf, stats, outseg, Nn);
}

extern "C" void kernel_launch(void* const* d_in, const int* in_sizes, int n_in,
                              void* d_out, int out_size, void* d_ws, size_t ws_size,
                              hipStream_t stream) {
    const float*     x  = (const float*)d_in[0];
    const long long* ei = (const long long*)d_in[1];
    const float* W1 = (const float*)d_in[2];  const float* b1 = (const float*)d_in[3];
    const float* W2 = (const float*)d_in[4];  const float* b2 = (const float*)d_in[5];
    const float* W3 = (const float*)d_in[6];  const float* b3 = (const float*)d_in[7];
    const float* gw1 = (const float*)d_in[8];  const float* gb1 = (const float*)d_in[9];
    const float* gw2 = (const float*)d_in[10]; const float* gb2 = (const float*)d_in[11];
    const float* gw3 = (const float*)d_in[12]; const float* gb3 = (const float*)d_in[13];

    const int Nn = in_sizes[0] / HD;      // 50000
    const int Ee = in_sizes[1] / 2;       // 800000
    const long long* srcp = ei;
    const long long* dstp = ei + Ee;
    const size_t NH = (size_t)Nn * HD;

    float* ws   = (float*)d_ws;
    float* h    = ws;                 // [N,H] gemm output
    float* bufA = ws + NH;            // [N,H] x1 / x3
    float* bufB = ws + 2 * NH;        // [N,H] x2
    float* dinv = ws + 3 * NH;        // [N]   deg -> dinv (in place)
    float* gate = dinv + Nn;          // [N]
    float* stats = gate + Nn;         // [2]
    float* out  = (float*)d_out;      // [3*H]

    zero_kernel<<<(out_size + 255) / 256, 256, 0, stream>>>(out, out_size);

    // degree + dinv (shared across the three layers)
    init_deg<<<(Nn + 255) / 256, 256, 0, stream>>>(dinv, Nn);
    count_deg<<<(Ee + 255) / 256, 256, 0, stream>>>(dstp, dinv, Ee);
    rsqrt_deg<<<(Nn + 255) / 256, 256, 0, stream>>>(dinv, Nn);

    // layer 1: x -> bufA,  pool -> out[0:128]
    run_layer(x,    W1, b1, gw1, gb1, srcp, dstp, dinv, h, bufA, gate, stats, out,        Nn, Ee, stream);
    // layer 2: bufA -> bufB, pool -> out[128:256]
    run_layer(bufA, W2, b2, gw2, gb2, srcp, dstp, dinv, h, bufB, gate, stats, out + HD,   Nn, Ee, stream);
    // layer 3: bufB -> bufA, pool -> out[256:384]
    run_layer(bufB, W3, b3, gw3, gb3, srcp, dstp, dinv, h, bufA, gate, stats, out + 2*HD, Nn, Ee, stream);
}